// GCN_one_14276471292070
// MI455X (gfx1250) — compile-verified
//
#include <hip/hip_runtime.h>

typedef __attribute__((ext_vector_type(2))) float v2f;
typedef __attribute__((ext_vector_type(8))) float v8f;

#define NFEAT 128
#define NHID  64

// ---------------------------------------------------------------------------
// Kernel 1: zero the accumulator (d_out) -- harness poisons it with 0xAA.
// ---------------------------------------------------------------------------
__global__ void gcn_zero_kernel(float* __restrict__ out, int n) {
    int i = blockIdx.x * blockDim.x + threadIdx.x;
    if (i < n) out[i] = 0.0f;
}

// ---------------------------------------------------------------------------
// Kernel 2: support = x @ W via V_WMMA_F32_16X16X4_F32.
// One wave32 per 16x16 output tile. K-loop: 32 WMMA ops (K=128, 4 per op).
//   A (16x4 f32): lanes 0-15 -> M=lane, VGPR0/1 = K=k0,k0+1
//                 lanes 16-31 -> M=lane-16, VGPR0/1 = K=k0+2,k0+3
//   B (4x16 f32): lane&15 = N, same K half-split as A.
//   C/D (16x16 f32, 8 VGPRs): row = m_base + j + 8*(lane>=16), col = lane&15.
// Block = 128 threads = 4 waves; grid = 6250 blocks -> 25000 tiles exactly.
// ---------------------------------------------------------------------------
__global__ void gcn_gemm_wmma_kernel(const float* __restrict__ x,
                                     const float* __restrict__ W,
                                     float* __restrict__ support) {
    const int wave   = blockIdx.x * 4 + (threadIdx.x >> 5);
    const int lane   = threadIdx.x & 31;
    const int tile_m = wave >> 2;         // 0..6249
    const int tile_n = wave & 3;          // 0..3
    const int m_base = tile_m * 16;
    const int n_base = tile_n * 16;
    const int half   = lane >> 4;         // 0 or 1
    const int l16    = lane & 15;
    const int klo    = half * 2;          // K sub-offset for this half-wave

    const float* __restrict__ xrow = x + (size_t)(m_base + l16) * NFEAT + klo;
    const float* __restrict__ wcol = W + n_base + l16;

    v8f c = {};
    #pragma unroll 8
    for (int k0 = 0; k0 < NFEAT; k0 += 4) {
        v2f a = *(const v2f*)(xrow + k0);          // global_load_b64, 8B aligned
        v2f b;
        b.x = wcol[(size_t)(k0 + klo) * NHID];
        b.y = wcol[(size_t)(k0 + klo + 1) * NHID];
        // 8 args: (neg_a, A, neg_b, B, c_mod, C, reuse_a, reuse_b)
        c = __builtin_amdgcn_wmma_f32_16x16x4_f32(false, a, false, b,
                                                  (short)0, c, false, false);
    }

    float* __restrict__ outp = support + (size_t)(m_base + half * 8) * NHID
                                       + n_base + l16;
    #pragma unroll
    for (int j = 0; j < 8; ++j) {
        outp[(size_t)j * NHID] = c[j];
    }
}

// ---------------------------------------------------------------------------
// Kernel 3: SpMM scatter-add. One wave32 per edge (grid-stride persistent
// waves). Lane l owns features [2l, 2l+1]: one b64 gather from support,
// two global_atomic_add_f32 into out. Working set (~51 MB) lives in the
// 192 MB L2, so the random traffic stays on-chip. Prefetch the next edge's
// support row to hide gather latency.
// ---------------------------------------------------------------------------
__global__ void gcn_spmm_kernel(const int* __restrict__ erow,
                                const int* __restrict__ ecol,
                                const float* __restrict__ eval,
                                const float* __restrict__ support,
                                float* __restrict__ out, int nE) {
    const int lane   = threadIdx.x & 31;
    const int wave   = (blockIdx.x * blockDim.x + threadIdx.x) >> 5;
    const int nWaves = (gridDim.x * blockDim.x) >> 5;

    for (int e = wave; e < nE; e += nWaves) {
        const int   r = erow[e];
        const int   c = ecol[e];
        const float v = eval[e];

        const int en = e + nWaves;
        if (en < nE) {
            const int cn = ecol[en];
            // global_prefetch_b8 of next gather target
            __builtin_prefetch(support + (size_t)cn * NHID + lane * 2, 0, 0);
        }

        const float2 s = ((const float2*)(support + (size_t)c * NHID))[lane];
        float* dst = out + (size_t)r * NHID + lane * 2;
        atomicAdd(dst,     v * s.x);
        atomicAdd(dst + 1, v * s.y);
    }
}

// ---------------------------------------------------------------------------
// Kernel 4: h = relu(agg + b); inverted dropout (p=0.5) with a fixed
// deterministic hash-based keep mask (stand-in for jax threefry key 42).
// ---------------------------------------------------------------------------
__global__ void gcn_epilogue_kernel(float* __restrict__ out,
                                    const float* __restrict__ bias, int n) {
    int i = blockIdx.x * blockDim.x + threadIdx.x;
    if (i >= n) return;
    float h = out[i] + bias[i & (NHID - 1)];
    h = fmaxf(h, 0.0f);
    // deterministic per-element keep bit, keyed by 42
    unsigned u = (unsigned)i * 0x9E3779B9u + 42u;
    u ^= u >> 16; u *= 0x7feb352du;
    u ^= u >> 15; u *= 0x846ca68bu;
    u ^= u >> 16;
    out[i] = (u & 1u) ? h * 2.0f : 0.0f;   // h / keep, keep = 0.5
}

// ---------------------------------------------------------------------------
extern "C" void kernel_launch(void* const* d_in, const int* in_sizes, int n_in,
                              void* d_out, int out_size, void* d_ws, size_t ws_size,
                              hipStream_t stream) {
    const float* x    = (const float*)d_in[0];   // [N, 128]
    const int*   erow = (const int*)  d_in[1];   // [E]
    const int*   ecol = (const int*)  d_in[2];   // [E]
    const float* eval = (const float*)d_in[3];   // [E]
    const float* W    = (const float*)d_in[4];   // [128, 64]
    const float* bias = (const float*)d_in[5];   // [64]
    float* out        = (float*)d_out;           // [N, 64]
    float* support    = (float*)d_ws;            // [N, 64] scratch

    const int N  = in_sizes[0] / NFEAT;          // 100000
    const int E  = in_sizes[1];                  // 1600000
    const int nO = N * NHID;                     // 6400000

    // 1) zero accumulator
    gcn_zero_kernel<<<(nO + 255) / 256, 256, 0, stream>>>(out, nO);

    // 2) support = x @ W  (one wave per 16x16 tile; 4 waves / block)
    const int mTiles = N / 16;                   // 6250
    gcn_gemm_wmma_kernel<<<mTiles, 128, 0, stream>>>(x, W, support);

    // 3) scatter-add over edges (persistent waves, 32768 waves)
    gcn_spmm_kernel<<<4096, 256, 0, stream>>>(erow, ecol, eval, support, out, E);

    // 4) bias + relu + dropout
    gcn_epilogue_kernel<<<(nO + 255) / 256, 256, 0, stream>>>(out, bias, nO);
}